// CARAFE3d_59992103190568
// MI455X (gfx1250) — compile-verified
//
#include <hip/hip_runtime.h>

// CARAFE3d for MI455X (gfx1250): f32 WMMA 16x16x4 end-to-end.
// x:[1,64,32,32,32] -> compress GEMM -> encode 3^3 conv (implicit GEMM, K=1728,
// LDS-staged B halo + async-to-LDS fill) -> fused pixelshuffle+softmax+
// reassembly+projection -> out:[1,32,64,64,64]

typedef __attribute__((ext_vector_type(2))) float v2f;
typedef __attribute__((ext_vector_type(8))) float v8f;

#define DHW 32768   // 32*32*32 source voxels
#define NV  262144  // 64*64*64 output voxels

__device__ __forceinline__ v8f wmma_f32(v2f a, v2f b, v8f c) {
  // D = A(16x4) * B(4x16) + C, full f32.
  return __builtin_amdgcn_wmma_f32_16x16x4_f32(false, a, false, b, (short)0, c,
                                               false, false);
}

__device__ __forceinline__ int clamp31(int v) {
  return v < 0 ? 0 : (v > 31 ? 31 : v);
}

// ---------------------------------------------------------------------------
// Kernel 1: cmid[64][32768] = w_compress[64x64] * x[64][32768] + b_compress
// ---------------------------------------------------------------------------
__global__ __launch_bounds__(256) void k_compress(const float* __restrict__ x,
                                                  const float* __restrict__ wc,
                                                  const float* __restrict__ bc,
                                                  float* __restrict__ cmid) {
  const int lane = threadIdx.x & 31;
  const int wave = threadIdx.x >> 5;
  const int row = lane & 15;
  const int hi  = lane >> 4;
  const int flat  = blockIdx.x * 8 + wave;
  const int ntile = flat >> 2;
  const int mtile = flat & 3;
  const int nbase = ntile * 16;
  const int mbase = mtile * 16;

  v8f acc = {};
#pragma unroll
  for (int ks = 0; ks < 16; ++ks) {
    const int ci0 = ks * 4 + hi * 2;
    v2f a = *(const v2f*)(wc + (mbase + row) * 64 + ci0);
    v2f b;
    b.x = x[ci0 * DHW + nbase + row];
    b.y = x[(ci0 + 1) * DHW + nbase + row];
    acc = wmma_f32(a, b, acc);
  }
#pragma unroll
  for (int v = 0; v < 8; ++v) {
    const int c = mbase + v + hi * 8;
    cmid[c * DHW + nbase + row] = acc[v] + bc[c];
  }
}

// ---------------------------------------------------------------------------
// Kernel 1b: transpose w_encode [216,64,3,3,3] -> weT[tap][m(224)][ci(64)]
// so encode A-fragments are contiguous b64 loads. Rows >=216 are zero.
// ---------------------------------------------------------------------------
#define WET_TOTAL (27 * 224 * 64)
__global__ __launch_bounds__(256) void k_wtrans(const float* __restrict__ we,
                                                float* __restrict__ weT) {
  const int idx = blockIdx.x * 256 + threadIdx.x;
  if (idx >= WET_TOTAL) return;
  const int tap = idx / (224 * 64);
  const int r   = idx % (224 * 64);
  const int m = r >> 6, ci = r & 63;
  weT[idx] = (m < 216) ? we[(m * 64 + ci) * 27 + tap] : 0.0f;
}

// ---------------------------------------------------------------------------
// Kernel 2: enc[216][32768] = conv3x3x3(cmid, w_encode) + b_encode
// One workgroup (7 waves) per 16-voxel n-tile. The zero-padded B halo
// [3z][3y][18x][64ci] is staged once in LDS via global_load_async_to_lds_b32
// (padding baked in at fill time), then every wave covers 2 of the 14
// M-tiles: inner loop = ds_load_b64(B, imm offset) + global_load_b64(A from
// tap-major weT) + v_wmma_f32_16x16x4_f32. 864 WMMAs per wave.
// ---------------------------------------------------------------------------
#define BSTRIDE 66  // 64 ci + 2 pad floats -> b64 lanes hit distinct banks
__global__ __launch_bounds__(224) void k_encode(const float* __restrict__ cmid,
                                                const float* __restrict__ weT,
                                                const float* __restrict__ be,
                                                float* __restrict__ enc) {
  __shared__ float blds[162 * BSTRIDE];  // 162 halo positions x 64 ci (+pad)
  const int tid  = threadIdx.x;
  const int lane = tid & 31;
  const int wave = tid >> 5;
  const int row = lane & 15;
  const int hi  = lane >> 4;

  const int nb = blockIdx.x * 16;
  const int xb = (blockIdx.x & 1) * 16;
  const int y  = (blockIdx.x >> 1) & 31;
  const int z  = blockIdx.x >> 6;

  // ---- stage B halo: blds[(zz*3+yy)*18+xx][ci] = cmid[ci][z+zz-1][y+yy-1][xb+xx-1]
  const unsigned lds_base = (unsigned)(size_t)&blds[0];  // LDS byte offset
  for (int e = tid; e < 162 * 64; e += 224) {
    const int ci  = e / 162;
    const int pos = e - ci * 162;
    const int zz = pos / 54;
    const int yy = (pos % 54) / 18;
    const int xx = pos % 18;
    const int sz = z + zz - 1, sy = y + yy - 1, sx = xb + xx - 1;
    const bool ok = (sz >= 0) & (sz < 32) & (sy >= 0) & (sy < 32) &
                    (sx >= 0) & (sx < 32);
    const unsigned loff = lds_base + (unsigned)(pos * BSTRIDE + ci) * 4u;
    if (ok) {
      const unsigned long long ga =
          (unsigned long long)(size_t)(cmid + ci * DHW + (sz * 1024 + sy * 32 + sx));
      asm volatile("global_load_async_to_lds_b32 %0, %1, off"
                   :: "v"(loff), "v"(ga)
                   : "memory");
    } else {
      blds[pos * BSTRIDE + ci] = 0.0f;  // zero padding baked in
    }
  }
  asm volatile("s_wait_asynccnt 0" ::: "memory");
  __syncthreads();

  // ---- compute: wave handles m-tiles {2*wave, 2*wave+1} (14 of 14 total)
#pragma unroll
  for (int mt = 0; mt < 2; ++mt) {
    const int mbase = (wave * 2 + mt) * 16;
    v8f acc = {};
    for (int tap = 0; tap < 27; ++tap) {
      const int zz = tap / 9;
      const int yy = (tap / 3) % 3;
      const int xx0 = tap % 3;
      const float* bp = blds + ((zz * 3 + yy) * 18 + xx0 + row) * BSTRIDE;
      const float* ap = weT + (tap * 224 + mbase + row) * 64;
#pragma unroll
      for (int ks = 0; ks < 16; ++ks) {
        const int ci0 = ks * 4 + hi * 2;
        v2f a = *(const v2f*)(ap + ci0);  // global_load_b64 (L2-resident weT)
        v2f b = *(const v2f*)(bp + ci0);  // ds_load_b64, immediate offset
        acc = wmma_f32(a, b, acc);
      }
    }
#pragma unroll
    for (int v = 0; v < 8; ++v) {
      const int c = mbase + v + hi * 8;
      if (c < 216) enc[c * DHW + nb + row] = acc[v] + be[c];
    }
  }
}

// ---------------------------------------------------------------------------
// Kernel 3 (fused): pixelshuffle + softmax(27) + edge-padded reassembly +
// 1x1x1 projection as WMMA GEMM [32x64]x[64x16]. One wave per 16 voxels.
// ---------------------------------------------------------------------------
__global__ __launch_bounds__(256) void k_upsample(const float* __restrict__ x,
                                                  const float* __restrict__ enc,
                                                  const float* __restrict__ wp,
                                                  float* __restrict__ out) {
  const int lane = threadIdx.x & 31;
  const int wave = threadIdx.x >> 5;
  const int row = lane & 15;
  const int hi  = lane >> 4;
  const int tile = blockIdx.x * 8 + wave;
  const int Xb = (tile & 3) * 16;
  const int Y  = (tile >> 2) & 63;
  const int Z  = tile >> 8;
  const int X  = Xb + row;

  const int d = Z >> 1, h = Y >> 1, wq = X >> 1;
  const int sub  = (Z & 1) * 4 + (Y & 1) * 2 + (X & 1);
  const int dvox = d * 1024 + h * 32 + wq;

  float wk[27];
  float mx = -1e30f;
#pragma unroll
  for (int k = 0; k < 27; ++k) {
    wk[k] = enc[(k * 8 + sub) * DHW + dvox];
    mx = fmaxf(mx, wk[k]);
  }
  float sum = 0.0f;
#pragma unroll
  for (int k = 0; k < 27; ++k) {
    wk[k] = __expf(wk[k] - mx);
    sum += wk[k];
  }
  const float inv = 1.0f / sum;

  float fl[32];
#pragma unroll
  for (int j = 0; j < 32; ++j) fl[j] = 0.0f;
  const float* xh = x + hi * 2 * DHW;
#pragma unroll
  for (int t = 0; t < 27; ++t) {
    const int kd = t / 9, kh = (t / 3) % 3, kw = t % 3;
    const int off = clamp31(d + kd - 1) * 1024 + clamp31(h + kh - 1) * 32 +
                    clamp31(wq + kw - 1);  // replicate padding
    const float wt = wk[t] * inv;
    const float* p = xh + off;
#pragma unroll
    for (int j = 0; j < 32; ++j) {
      const int cc = 4 * (j >> 1) + (j & 1);
      fl[j] = fmaf(wt, p[cc * DHW], fl[j]);
    }
  }

  v8f acc0 = {}, acc1 = {};
#pragma unroll
  for (int s = 0; s < 16; ++s) {
    const int ci0 = s * 4 + hi * 2;
    v2f b;
    b.x = fl[2 * s];
    b.y = fl[2 * s + 1];
    v2f a0 = *(const v2f*)(wp + row * 64 + ci0);
    v2f a1 = *(const v2f*)(wp + (16 + row) * 64 + ci0);
    acc0 = wmma_f32(a0, b, acc0);
    acc1 = wmma_f32(a1, b, acc1);
  }

  const int vox = Z * 4096 + Y * 64 + X;
#pragma unroll
  for (int v = 0; v < 8; ++v) {
    const int c = v + hi * 8;
    out[c * NV + vox] = acc0[v];
    out[(16 + c) * NV + vox] = acc1[v];
  }
}

// ---------------------------------------------------------------------------
extern "C" void kernel_launch(void* const* d_in, const int* in_sizes, int n_in,
                              void* d_out, int out_size, void* d_ws,
                              size_t ws_size, hipStream_t stream) {
  const float* x  = (const float*)d_in[0];  // [64,32,32,32]
  const float* wc = (const float*)d_in[1];  // [64,64,1,1,1]
  const float* bc = (const float*)d_in[2];  // [64]
  const float* we = (const float*)d_in[3];  // [216,64,3,3,3]
  const float* be = (const float*)d_in[4];  // [216]
  const float* wp = (const float*)d_in[5];  // [32,64,1,1,1]
  float* out = (float*)d_out;               // [32,64,64,64]

  float* cmid = (float*)d_ws;            // 64  * 32768 f32 (8 MB)
  float* enc  = cmid + 64 * DHW;         // 216 * 32768 f32 (28.3 MB)
  float* weT  = enc + 216 * DHW;         // 27*224*64 f32 (1.5 MB)

  k_compress<<<1024, 256, 0, stream>>>(x, wc, bc, cmid);
  k_wtrans<<<(WET_TOTAL + 255) / 256, 256, 0, stream>>>(we, weT);
  k_encode<<<2048, 224, 0, stream>>>(cmid, weT, be, enc);
  k_upsample<<<2048, 256, 0, stream>>>(x, enc, wp, out);
}